// Distance_56650618634565
// MI455X (gfx1250) — compile-verified
//
#include <hip/hip_runtime.h>
#include <hip/hip_bf16.h>
#include <math.h>

typedef __attribute__((ext_vector_type(2))) float v2f;
typedef __attribute__((ext_vector_type(4))) float v4f;
typedef __attribute__((ext_vector_type(8))) float v8f;
typedef __attribute__((ext_vector_type(4))) int   v4i;

#define EPS 1e-8f

#if __has_builtin(__builtin_amdgcn_global_load_async_to_lds_b128)
#define ASYNC_LDS 1
#else
#define ASYNC_LDS 0
#endif

// ---------------------------------------------------------------------------
// Kernel 1: per-row sum of squares (one wave32 per row, shuffle reduction)
// ---------------------------------------------------------------------------
__global__ void rownorm_kernel(const float* __restrict__ src,
                               float* __restrict__ dst,
                               int rows, int d) {
    int wave = (blockIdx.x * blockDim.x + threadIdx.x) >> 5;
    int lane = threadIdx.x & 31;
    if (wave >= rows) return;
    const float* p = src + (size_t)wave * d;
    float s = 0.0f;
    for (int i = lane; i < d; i += 32) {
        float v = p[i];
        s = fmaf(v, v, s);
    }
    #pragma unroll
    for (int off = 16; off > 0; off >>= 1)
        s += __shfl_xor(s, off, 32);
    if (lane == 0) dst[wave] = s;
}

// ---------------------------------------------------------------------------
// Kernel 2: fused L1 (packed/dual VALU) + dot GEMM (fp32 WMMA).
//   block = 256 threads = 8 wave32, each wave owns a 16x16 (M x N) tile,
//   block tile = 128 features x 16 centroids. D staged through LDS in
//   chunks of 32 via GLOBAL_LOAD_ASYNC_TO_LDS_B128 (ASYNCcnt) when
//   available; WMMA consumes k=4 slices (V_WMMA_F32_16X16X4_F32) on the
//   matrix pipe while the VALU pipes accumulate |f-c|.
// ---------------------------------------------------------------------------
#define KSTAGE 32
#define FSTR   36   // LDS row stride in floats (144B: 16B aligned, conflict pad)
#define CSTR   36

__global__ __launch_bounds__(256)
void fused_dist_kernel(const float* __restrict__ features,
                       const float* __restrict__ centroids,
                       const float* __restrict__ fsq,
                       const float* __restrict__ csq,
                       float* __restrict__ out,   // [3][N][C] flattened
                       int nRows, int nC, int d) {
    __shared__ __align__(16) float fLds[128 * FSTR];
    __shared__ __align__(16) float cLds[16 * CSTR];

    const int tid    = threadIdx.x;
    const int waveId = tid >> 5;          // 0..7
    const int lane   = tid & 31;
    const int lm     = lane & 15;         // N within tile / M row for A-frag
    const int hi     = lane >> 4;         // half-wave selector
    const int mW     = waveId << 4;       // wave's M offset inside block tile

    const int c0 = blockIdx.x << 4;       // centroid tile base (16 wide)
    const int m0 = blockIdx.y << 7;       // feature tile base (128 tall)

    const float scale = 1.0f / sqrtf((float)d);

    v8f   accd = {};                      // WMMA dot accumulator (16x16 f32 C/D)
    float acc1[8] = {};                   // L1 accumulator, same (M,N) mapping

    // per-lane LDS bases (constant across the k-loop -> immediate ds offsets)
    const float* aBase  = &fLds[(mW + lm) * FSTR + 2 * hi]; // A[M=lm][k+2hi..]
    const float* bBase  = &cLds[lm * CSTR + 2 * hi];        // B[k+2hi..][N=lm]
    const float* cvBase = &cLds[lm * CSTR];                 // centroid row lm
    const float* fBase  = &fLds[(mW + 8 * hi) * FSTR];      // L1 rows base

    for (int k0 = 0; k0 < d; k0 += KSTAGE) {
#if ASYNC_LDS
        // ---- async global -> LDS staging (no VGPR round-trip) -------------
        #pragma unroll
        for (int i = 0; i < 4; ++i) {
            int lin = tid + (i << 8);            // 0..1023
            int row = lin >> 3;                  // 0..127
            int c4  = (lin & 7) << 2;            // 0,4,...,28
            int m   = min(m0 + row, nRows - 1);
            __builtin_amdgcn_global_load_async_to_lds_b128(
                (v4i*)&features[(size_t)m * d + k0 + c4],
                (__attribute__((address_space(3))) v4i*)&fLds[row * FSTR + c4],
                0, 0);
        }
        if (tid < 128) {
            int row = tid >> 3;                  // 0..15
            int c4  = (tid & 7) << 2;
            int ci  = min(c0 + row, nC - 1);
            __builtin_amdgcn_global_load_async_to_lds_b128(
                (v4i*)&centroids[(size_t)ci * d + k0 + c4],
                (__attribute__((address_space(3))) v4i*)&cLds[row * CSTR + c4],
                0, 0);
        }
#if __has_builtin(__builtin_amdgcn_s_wait_asynccnt)
        __builtin_amdgcn_s_wait_asynccnt(0);
#else
        asm volatile("s_wait_asynccnt 0x0" ::: "memory");
#endif
#else
        // ---- fallback: register staging -----------------------------------
        #pragma unroll
        for (int i = 0; i < 4; ++i) {
            int lin = tid + (i << 8);
            int row = lin >> 3;
            int c4  = (lin & 7) << 2;
            int m   = min(m0 + row, nRows - 1);
            *(v4f*)&fLds[row * FSTR + c4] =
                *(const v4f*)&features[(size_t)m * d + k0 + c4];
        }
        if (tid < 128) {
            int row = tid >> 3;
            int c4  = (tid & 7) << 2;
            int ci  = min(c0 + row, nC - 1);
            *(v4f*)&cLds[row * CSTR + c4] =
                *(const v4f*)&centroids[(size_t)ci * d + k0 + c4];
        }
#endif
        // prefetch next feature stage into cache (global_prefetch_b8)
        if (k0 + KSTAGE < d) {
            int row = tid >> 3;
            int m   = min(m0 + row, nRows - 1);
            __builtin_prefetch(&features[(size_t)m * d + k0 + KSTAGE], 0, 0);
        }
        __syncthreads();

        // ---- compute: per k4 slice, 1 fp32 WMMA + 8 rows of L1 VALU -------
        #pragma unroll
        for (int kk = 0; kk < KSTAGE; kk += 4) {
            // WMMA fragments straight from aligned ds_load_b64
            v2f a = *(const v2f*)(aBase + kk);
            v2f b = *(const v2f*)(bBase + kk);
            // centroid slice for this lane's N column (L1 path)
            v4f cv = *(const v4f*)(cvBase + kk);

            accd = __builtin_amdgcn_wmma_f32_16x16x4_f32(
                false, a, false, b, (short)0, accd, false, false);

            // L1: rows M = mW + 8*hi + r, column N = lm
            #pragma unroll
            for (int r = 0; r < 8; ++r) {
                v4f fv = *(const v4f*)(fBase + r * FSTR + kk);
                acc1[r] += fabsf(fv.x - cv.x) + fabsf(fv.y - cv.y) +
                           fabsf(fv.z - cv.z) + fabsf(fv.w - cv.w);
            }
        }
        __syncthreads();
    }

    // ---- epilogue: l1 / l2 / cos ------------------------------------------
    const int n = c0 + lm;
    const float cs = csq[min(n, nC - 1)];
    const float cn = fmaxf(sqrtf(cs), EPS);
    const size_t NC = (size_t)nRows * nC;
    float* out_l1  = out;
    float* out_l2  = out + NC;
    float* out_cos = out + 2 * NC;

    #pragma unroll
    for (int r = 0; r < 8; ++r) {
        int m = m0 + mW + 8 * hi + r;
        if (n < nC && m < nRows) {
            float fs  = fsq[m];
            float dot = accd[r];
            size_t o  = (size_t)m * nC + n;
            out_l1[o] = acc1[r] * scale;
            float sq  = fmaxf(fs + cs - 2.0f * dot, 0.0f);
            out_l2[o] = sqrtf(sq) * scale;
            float fn  = fmaxf(sqrtf(fs), EPS);
            out_cos[o] = (dot / (fn * cn)) * scale;
        }
    }
}

// ---------------------------------------------------------------------------
extern "C" void kernel_launch(void* const* d_in, const int* in_sizes, int n_in,
                              void* d_out, int out_size, void* d_ws, size_t ws_size,
                              hipStream_t stream) {
    const float* features  = (const float*)d_in[0];
    const float* centroids = (const float*)d_in[1];
    float* out = (float*)d_out;

    const int D = 512;
    const int N = in_sizes[0] / D;
    const int C = in_sizes[1] / D;

    float* fsq = (float*)d_ws;       // N floats
    float* csq = fsq + N;            // C floats

    // row norms: one wave32 per row, 8 rows per 256-thread block
    {
        int blocks = (N + 7) / 8;
        rownorm_kernel<<<blocks, 256, 0, stream>>>(features, fsq, N, D);
    }
    {
        int blocks = (C + 7) / 8;
        rownorm_kernel<<<blocks, 256, 0, stream>>>(centroids, csq, C, D);
    }

    dim3 grid((C + 15) / 16, (N + 127) / 128);
    fused_dist_kernel<<<grid, 256, 0, stream>>>(features, centroids,
                                                fsq, csq, out, N, C, D);
}